// SymmetricContraction_89979564851557
// MI455X (gfx1250) — compile-verified
//
#include <hip/hip_runtime.h>

typedef __attribute__((ext_vector_type(2))) float v2f;
typedef __attribute__((ext_vector_type(8))) float v8f;

#define NB 1024
#define NC 128
#define NL 16
#define NE 10

#define P3_0E 23
#define P2_0E 4
#define P3_1O 37
#define P2_1O 6

#define M0E 256   // (w,x) rows for 0e
#define M1O 768   // (w,x,v) rows for 1o

// ---- workspace layout (float offsets) ----
#define OFF_A3_0E 0ll
#define SZ_A3_0E  (10ll*128*256*16)      // 5,242,880
#define OFF_A3_1O (OFF_A3_0E + SZ_A3_0E)
#define SZ_A3_1O  (10ll*128*768*16)      // 15,728,640
#define OFF_A2_0E (OFF_A3_1O + SZ_A3_1O)
#define SZ_A2_0E  (10ll*128*256)
#define OFF_A2_1O (OFF_A2_0E + SZ_A2_0E)
#define SZ_A2_1O  (10ll*128*768)
#define OFF_A1_0E (OFF_A2_1O + SZ_A2_1O)
#define SZ_A1_0E  (10ll*128*16)
#define OFF_A1_1O (OFF_A1_0E + SZ_A1_0E)
#define SZ_A1_1O  (10ll*128*48)
#define OFF_PERM  (OFF_A1_1O + SZ_A1_1O) // int region: perm[E*B] then counts[E]
// total ~89.5 MB + 41KB of d_ws

// LDS: A3 rows [1024][18] (pad 16->18 floats: 16-lane row stride hits all banks),
// A2[1024], A1[64], per-wave x-tile sXv[8][256]
#define SA3_FLOATS  (1024*18)
#define SA2_FLOATS  1024
#define SA1_FLOATS  64
#define SXV_FLOATS  (8*256)
#define SMEM_FLOATS (SA3_FLOATS + SA2_FLOATS + SA1_FLOATS + SXV_FLOATS)

__global__ void k_zero(int* __restrict__ counts) {
    if (threadIdx.x < NE) counts[threadIdx.x] = 0;
}

// bucket nodes by element (argmax of one-hot row)
__global__ void k_perm(const float* __restrict__ y, int* __restrict__ perm,
                       int* __restrict__ counts) {
    int b = blockIdx.x * blockDim.x + threadIdx.x;
    if (b >= NB) return;
    float best = y[b * NE];
    int e = 0;
    #pragma unroll
    for (int j = 1; j < NE; ++j) {
        float v = y[b * NE + j];
        if (v > best) { best = v; e = j; }
    }
    int pos = atomicAdd(&counts[e], 1);
    perm[e * NB + pos] = b;
}

// Precompute A3[e,c,m,0:16] = sum_k U3[m,i,k] W3[e,k,c]  (and A2, A1)
__global__ void k_precompute(const float* __restrict__ U1_0e, const float* __restrict__ U2_0e,
                             const float* __restrict__ U3_0e,
                             const float* __restrict__ W1_0e, const float* __restrict__ W2_0e,
                             const float* __restrict__ W3_0e,
                             const float* __restrict__ U1_1o, const float* __restrict__ U2_1o,
                             const float* __restrict__ U3_1o,
                             const float* __restrict__ W1_1o, const float* __restrict__ W2_1o,
                             const float* __restrict__ W3_1o,
                             float* __restrict__ ws) {
    long long idx = (long long)blockIdx.x * 256 + threadIdx.x;
    const long long N0 = (long long)NE * NC * M0E;
    const long long N1 = (long long)NE * NC * M1O;
    if (idx < N0) {
        int c = (int)(idx % NC);
        long long r = idx / NC;
        int m = (int)(r % M0E);
        int e = (int)(r / M0E);
        float acc[16];
        #pragma unroll
        for (int i = 0; i < 16; ++i) acc[i] = 0.f;
        for (int k = 0; k < P3_0E; ++k) {
            float wv = W3_0e[(e * P3_0E + k) * NC + c];
            #pragma unroll
            for (int i = 0; i < 16; ++i)
                acc[i] += U3_0e[(m * 16 + i) * P3_0E + k] * wv;
        }
        float* dst = ws + OFF_A3_0E + (((long long)(e * NC + c) * M0E + m) << 4);
        #pragma unroll
        for (int q = 0; q < 4; ++q)
            ((float4*)dst)[q] = make_float4(acc[q*4], acc[q*4+1], acc[q*4+2], acc[q*4+3]);
        float a2 = 0.f;
        #pragma unroll
        for (int k = 0; k < P2_0E; ++k)
            a2 += U2_0e[m * P2_0E + k] * W2_0e[(e * P2_0E + k) * NC + c];
        ws[OFF_A2_0E + (long long)(e * NC + c) * M0E + m] = a2;
        if (m < 16)
            ws[OFF_A1_0E + (long long)(e * NC + c) * 16 + m] = U1_0e[m] * W1_0e[e * NC + c];
    } else if (idx < N0 + N1) {
        long long j = idx - N0;
        int c = (int)(j % NC);
        long long r = j / NC;
        int m = (int)(r % M1O);
        int e = (int)(r / M1O);
        float acc[16];
        #pragma unroll
        for (int i = 0; i < 16; ++i) acc[i] = 0.f;
        for (int k = 0; k < P3_1O; ++k) {
            float wv = W3_1o[(e * P3_1O + k) * NC + c];
            #pragma unroll
            for (int i = 0; i < 16; ++i)
                acc[i] += U3_1o[(m * 16 + i) * P3_1O + k] * wv;
        }
        float* dst = ws + OFF_A3_1O + (((long long)(e * NC + c) * M1O + m) << 4);
        #pragma unroll
        for (int q = 0; q < 4; ++q)
            ((float4*)dst)[q] = make_float4(acc[q*4], acc[q*4+1], acc[q*4+2], acc[q*4+3]);
        float a2 = 0.f;
        #pragma unroll
        for (int k = 0; k < P2_1O; ++k)
            a2 += U2_1o[m * P2_1O + k] * W2_1o[(e * P2_1O + k) * NC + c];
        ws[OFF_A2_1O + (long long)(e * NC + c) * M1O + m] = a2;
        if (m < 48)  // m here is the (w,x) tile index for A1_1o
            ws[OFF_A1_1O + (long long)(e * NC + c) * 48 + m] = U1_1o[m] * W1_1o[e * NC + c];
    }
}

__global__ void __launch_bounds__(256)
k_main(const float* __restrict__ x, const float* __restrict__ ws,
       const int* __restrict__ perm, const int* __restrict__ counts,
       float* __restrict__ out) {
    extern __shared__ float smem[];
    float* sA3 = smem;                       // [1024 rows][18]
    float* sA2 = sA3 + SA3_FLOATS;           // [1024]
    float* sA1 = sA2 + SA2_FLOATS;           // [64]
    float* sXv = sA1 + SA1_FLOATS;           // [8 waves][16 cols][16]

    const int c = blockIdx.x;
    const int e = blockIdx.y;
    const int tid = threadIdx.x;

    const float* g3a = ws + OFF_A3_0E + (long long)(e * NC + c) * (M0E * 16);
    const float* g3b = ws + OFF_A3_1O + (long long)(e * NC + c) * (M1O * 16);
    const float* g2a = ws + OFF_A2_0E + (long long)(e * NC + c) * M0E;
    const float* g2b = ws + OFF_A2_1O + (long long)(e * NC + c) * M1O;
    const float* g1a = ws + OFF_A1_0E + (long long)(e * NC + c) * 16;
    const float* g1b = ws + OFF_A1_1O + (long long)(e * NC + c) * 48;

    for (int i = tid; i < M0E * 16; i += 256) sA3[(i >> 4) * 18 + (i & 15)] = g3a[i];
    for (int i = tid; i < M1O * 16; i += 256) sA3[(256 + (i >> 4)) * 18 + (i & 15)] = g3b[i];
    if (tid < M0E) sA2[tid] = g2a[tid];
    for (int i = tid; i < M1O; i += 256) sA2[256 + i] = g2b[i];
    if (tid < 16) sA1[tid] = g1a[tid];
    else if (tid < 64) sA1[tid] = g1b[tid - 16];
    __syncthreads();

    const int lane = tid & 31;
    const int wvid = tid >> 5;
    const int half = lane >> 4;   // WMMA C-layout: lanes 0-15 rows 0-7, 16-31 rows 8-15
    const int col  = lane & 15;   // node column within tile
    const int cnt = counts[e];
    const int ntiles = (cnt + 15) >> 4;

    float* sxw = sXv + wvid * 256;              // this wave's x tile [16 cols][16]
    const float* sxcol = sxw + col * 16;        // this lane's column

    for (int nt = wvid; nt < ntiles; nt += 8) {
        const int nidx = nt * 16 + col;
        const bool valid = nidx < cnt;
        const int b = perm[e * NB + (valid ? nidx : nt * 16)];
        const float* xb = x + ((long long)b * NC + c) * NL;

        // this half's 8 x-values (rows half*8 .. half*8+7)
        float4 xlo = *(const float4*)(xb + half * 8);
        float4 xhi = *(const float4*)(xb + half * 8 + 4);
        float xm[8] = {xlo.x, xlo.y, xlo.z, xlo.w, xhi.x, xhi.y, xhi.z, xhi.w};

        // stage full column into LDS (lane pair covers all 16 values)
        ((float4*)(sxcol + half * 8))[0] = xlo;
        ((float4*)(sxcol + half * 8))[1] = xhi;
        __builtin_amdgcn_wave_barrier();   // LDS is in-order within a wave

        // B fragments: B[k,n]; lanes 0-15 hold k = 4kk..4kk+1, lanes 16-31 k = 4kk+2..4kk+3
        v2f bf[4];
        #pragma unroll
        for (int kk = 0; kk < 4; ++kk) {
            bf[kk].x = sxcol[kk * 4 + half * 2 + 0];
            bf[kk].y = sxcol[kk * 4 + half * 2 + 1];
        }

        // 4 tile groups: g=0 -> out0 (tiles 0..15), g=1..3 -> o1[g-1] (tiles 16g..16g+15)
        float res[4];
        #pragma unroll
        for (int g = 0; g < 4; ++g) {
            float accg = 0.f;
            #pragma unroll 4
            for (int tt = 0; tt < 16; ++tt) {
                const int t = g * 16 + tt;
                // acc init = A2 row (step 2 folded in); C layout rows split by half
                v8f acc;
                #pragma unroll
                for (int r = 0; r < 8; ++r) acc[r] = sA2[t * 16 + half * 8 + r];
                // D = A3tile(16x16) * X(16x16) + acc  via 4 chained K=4 WMMAs
                #pragma unroll
                for (int kk = 0; kk < 4; ++kk) {
                    const float* ap = sA3 + (t * 16 + col) * 18 + kk * 4 + half * 2;
                    v2f af; af.x = ap[0]; af.y = ap[1];
                    acc = __builtin_amdgcn_wmma_f32_16x16x4_f32(
                        false, af, false, bf[kk], (short)0, acc, false, false);
                }
                // step 3: contract tile rows with x; fold the two 8-row halves
                float s = 0.f;
                #pragma unroll
                for (int r = 0; r < 8; ++r) s += acc[r] * xm[r];
                s += __shfl_xor(s, 16, 32);
                // steps 4+5: (A1 + out2) * x[row], full weight from LDS (both halves)
                accg += (sA1[t] + s) * sxcol[tt];
            }
            res[g] = accg;
        }

        if (valid && half == 0) {
            float* o = out + (long long)b * (4 * NC);
            o[c] = res[0];                  // [B, C]   0e block
            o[NC + c * 3 + 0] = res[1];    // [B, 3C]  1o block, idx c*3+w
            o[NC + c * 3 + 1] = res[2];
            o[NC + c * 3 + 2] = res[3];
        }
    }
}

extern "C" void kernel_launch(void* const* d_in, const int* in_sizes, int n_in,
                              void* d_out, int out_size, void* d_ws, size_t ws_size,
                              hipStream_t stream) {
    const float* x     = (const float*)d_in[0];
    const float* y     = (const float*)d_in[1];
    const float* U1_0e = (const float*)d_in[2];
    const float* U2_0e = (const float*)d_in[3];
    const float* U3_0e = (const float*)d_in[4];
    const float* W1_0e = (const float*)d_in[5];
    const float* W2_0e = (const float*)d_in[6];
    const float* W3_0e = (const float*)d_in[7];
    const float* U1_1o = (const float*)d_in[8];
    const float* U2_1o = (const float*)d_in[9];
    const float* U3_1o = (const float*)d_in[10];
    const float* W1_1o = (const float*)d_in[11];
    const float* W2_1o = (const float*)d_in[12];
    const float* W3_1o = (const float*)d_in[13];

    float* ws   = (float*)d_ws;                 // needs ~90 MB of workspace
    int* perm   = (int*)(ws + OFF_PERM);
    int* counts = perm + NE * NB;
    float* out  = (float*)d_out;

    k_zero<<<1, 32, 0, stream>>>(counts);
    k_perm<<<NB / 256, 256, 0, stream>>>(y, perm, counts);

    const long long total = (long long)NE * NC * (M0E + M1O);   // 1,310,720
    k_precompute<<<(int)((total + 255) / 256), 256, 0, stream>>>(
        U1_0e, U2_0e, U3_0e, W1_0e, W2_0e, W3_0e,
        U1_1o, U2_1o, U3_1o, W1_1o, W2_1o, W3_1o, ws);

    const size_t smem = SMEM_FLOATS * sizeof(float);  // ~86 KB (gfx1250 LDS: 320KB/WGP)
    k_main<<<dim3(NC, NE), 256, smem, stream>>>(x, ws, perm, counts, out);
}